// Net_29403346109054
// MI455X (gfx1250) — compile-verified
//
#include <hip/hip_runtime.h>

typedef float v2f __attribute__((ext_vector_type(2)));
typedef float v8f __attribute__((ext_vector_type(8)));

#define NTOT   50000
#define DEG    16
#define NG     50
#define NN     1000
#define NC     64
#define HD     128
#define H2D    256
#define NEDGE  (NTOT * DEG)
#define BN_EPS 1e-5f

// ---------------------------------------------------------------------------
// Generic fp32 WMMA GEMM:  Y[M,N] = relu?( X[M,K] @ W[K,N] + bias[N] )
// One wave computes a 16x16 tile via V_WMMA_F32_16X16X4_F32, K-step = 4.
// Block = 256 threads = 8 waves covering 16 rows x 128 cols.
// grid = (M/16, N/128). M % 16 == 0, N % 128 == 0, K % 4 == 0 required.
// ---------------------------------------------------------------------------
__global__ void gemm_bias_wmma(const float* __restrict__ X,
                               const float* __restrict__ W,
                               const float* __restrict__ bias,
                               float* __restrict__ Y,
                               int M, int K, int N, int relu) {
  const int lane = threadIdx.x & 31;
  const int wave = threadIdx.x >> 5;
  const int row0 = blockIdx.x * 16;
  const int col0 = blockIdx.y * 128 + wave * 16;
  const int half = lane >> 4;   // 0: lanes 0-15, 1: lanes 16-31
  const int lm   = lane & 15;

  v8f acc = {};
  // A operand: lane (half==0) holds X[row0+lm, k..k+1]; half==1 holds k+2..k+3
  const float* Xr = X + (size_t)(row0 + lm) * K + 2 * half;
  // B operand: lane holds W[k + 2*half + {0,1}, col0+lm]
  const float* Wc = W + (size_t)(2 * half) * N + (col0 + lm);

  #pragma unroll 4
  for (int kb = 0; kb < K; kb += 4) {
    v2f a, b;
    a.x = Xr[kb];
    a.y = Xr[kb + 1];
    b.x = Wc[(size_t)kb * N];
    b.y = Wc[(size_t)(kb + 1) * N];
    acc = __builtin_amdgcn_wmma_f32_16x16x4_f32(
        /*neg_a=*/false, a, /*neg_b=*/false, b,
        /*c_mod=*/(short)0, acc, /*reuse_a=*/false, /*reuse_b=*/false);
  }

  const int col = col0 + lm;
  const float bv = bias[col];
  #pragma unroll
  for (int r = 0; r < 8; ++r) {
    const int row = row0 + r + 8 * half;   // C/D layout: VGPR r = M r / r+8
    float v = acc[r] + bv;
    if (relu) v = fmaxf(v, 0.0f);
    Y[(size_t)row * N + col] = v;
  }
}

// ---------------------------------------------------------------------------
// GCN aggregation:  out[i,:] = h[i,:] + sum_{e in [16i,16i+16)} w[e] * h[src[e],:]
// Exploits dst = repeat(arange(NTOT), DEG): node i owns edges [16i, 16i+16).
// One wave per node; each lane owns a float4 (128 feats / 32 lanes).
// ---------------------------------------------------------------------------
__global__ void agg_gcn_kernel(const float* __restrict__ h,
                               const float* __restrict__ w,
                               const int* __restrict__ src,
                               float* __restrict__ out) {
  const int lane = threadIdx.x & 31;
  const int wave = threadIdx.x >> 5;
  const int node = blockIdx.x * 8 + wave;   // grid = NTOT/8 exact

  const float4* selfp = (const float4*)(h + (size_t)node * HD);
  float4 acc = selfp[lane];
  const int base = node * DEG;
  #pragma unroll
  for (int e = 0; e < DEG; ++e) {
    const int   s  = src[base + e];
    const float we = w[base + e];
    const float4 hv = ((const float4*)(h + (size_t)s * HD))[lane];
    acc.x += we * hv.x; acc.y += we * hv.y;
    acc.z += we * hv.z; acc.w += we * hv.w;
  }
  ((float4*)(out + (size_t)node * HD))[lane] = acc;
}

// GIN aggregation: out = (1+eps[l]) * h[i] + sum w[e] * h[src[e]]
__global__ void agg_gin_kernel(const float* __restrict__ h,
                               const float* __restrict__ w,
                               const int* __restrict__ src,
                               const float* __restrict__ eps, int l,
                               float* __restrict__ out) {
  const int lane = threadIdx.x & 31;
  const int wave = threadIdx.x >> 5;
  const int node = blockIdx.x * 8 + wave;

  const float scale = 1.0f + eps[l];
  const float4* selfp = (const float4*)(h + (size_t)node * HD);
  float4 sv = selfp[lane];
  float4 acc = { scale * sv.x, scale * sv.y, scale * sv.z, scale * sv.w };
  const int base = node * DEG;
  #pragma unroll
  for (int e = 0; e < DEG; ++e) {
    const int   s  = src[base + e];
    const float we = w[base + e];
    const float4 hv = ((const float4*)(h + (size_t)s * HD))[lane];
    acc.x += we * hv.x; acc.y += we * hv.y;
    acc.z += we * hv.z; acc.w += we * hv.w;
  }
  ((float4*)(out + (size_t)node * HD))[lane] = acc;
}

// ---------------------------------------------------------------------------
// BatchNorm pass 1: per-column partial Σx, Σx² accumulated with atomics.
// blockDim.x == F; each block handles rowsPerBlock rows.
// ---------------------------------------------------------------------------
__global__ void bn_stats_kernel(const float* __restrict__ X,
                                float* __restrict__ sums,
                                float* __restrict__ sqs,
                                int rows, int F, int rowsPerBlock) {
  const int col = threadIdx.x;
  int r0 = blockIdx.x * rowsPerBlock;
  int r1 = r0 + rowsPerBlock; if (r1 > rows) r1 = rows;
  float s = 0.0f, q = 0.0f;
  for (int r = r0; r < r1; ++r) {
    const float v = X[(size_t)r * F + col];
    s += v; q += v * v;
  }
  atomicAdd(&sums[col], s);
  atomicAdd(&sqs[col], q);
}

// BatchNorm pass 2 (+ReLU): y = relu((x - m) / sqrt(var + eps))
__global__ void bn_apply_relu(const float* __restrict__ X,
                              float* __restrict__ Y,
                              const float* __restrict__ sums,
                              const float* __restrict__ sqs,
                              int rows, int F) {
  const size_t i = (size_t)blockIdx.x * blockDim.x + threadIdx.x;
  if (i >= (size_t)rows * F) return;
  const int col = (int)(i % F);
  const float inv = 1.0f / (float)rows;
  const float m = sums[col] * inv;
  const float v = sqs[col] * inv - m * m;
  const float y = (X[i] - m) * rsqrtf(v + BN_EPS);
  Y[i] = fmaxf(y, 0.0f);
}

__global__ void zero_f32(float* __restrict__ p, int n) {
  const int i = blockIdx.x * blockDim.x + threadIdx.x;
  if (i < n) p[i] = 0.0f;
}

// Cluster pooling: pooled[gid*C + labels[i], :] += concat(h_gcn[i], h_gin[i])
__global__ void pool_cluster(const float* __restrict__ hg,
                             const float* __restrict__ hm,
                             const int* __restrict__ labels,
                             float* __restrict__ pooled) {
  const int i = blockIdx.x;        // node
  const int f = threadIdx.x;       // 0..127
  const int seg = (i / NN) * NC + labels[i];
  atomicAdd(&pooled[(size_t)seg * H2D + f],      hg[(size_t)i * HD + f]);
  atomicAdd(&pooled[(size_t)seg * H2D + HD + f], hm[(size_t)i * HD + f]);
}

// Pooled adjacency: A[b, labels[src_g[e]], labels[dst[e]]] += 1  (dst[e] = e/16)
__global__ void build_A(const int* __restrict__ src_g,
                        const int* __restrict__ labels,
                        float* __restrict__ A) {
  const int e = blockIdx.x * blockDim.x + threadIdx.x;
  if (e >= NEDGE) return;
  const int d  = e / DEG;
  const int b  = d / NN;
  const int ls = labels[src_g[e]];
  const int ld = labels[d];
  atomicAdd(&A[((size_t)b * NC + ls) * NC + ld], 1.0f);
}

__global__ void set_diag(float* __restrict__ A) {
  A[((size_t)blockIdx.x * NC + threadIdx.x) * NC + threadIdx.x] = 1.0f;
}

// out[b,y,d] = hp[b,y,d] + sum_x A[b,x,y] * hp[b,x,d]   (einsum bxy,bxd->byd + id)
__global__ void adj_mm(const float* __restrict__ A,
                       const float* __restrict__ hp,
                       float* __restrict__ out) {
  const int b = blockIdx.x, y = blockIdx.y, d = threadIdx.x;  // block = 256
  float acc = hp[((size_t)b * NC + y) * H2D + d];
  const float* Ab = A + (size_t)b * NC * NC;
  #pragma unroll 8
  for (int x = 0; x < NC; ++x)
    acc += Ab[x * NC + y] * hp[((size_t)b * NC + x) * H2D + d];
  out[((size_t)b * NC + y) * H2D + d] = acc;
}

// logits[r, o] = X[r,:] @ cls_W[:,o] + cls_b[o]
__global__ void cls_logits(const float* __restrict__ X,
                           const float* __restrict__ Wc,
                           const float* __restrict__ bc,
                           float* __restrict__ logits, int rows) {
  const int t = blockIdx.x * blockDim.x + threadIdx.x;
  if (t >= rows * 10) return;
  const int r = t / 10, o = t % 10;
  float acc = bc[o];
  for (int k = 0; k < H2D; ++k)
    acc += X[(size_t)r * H2D + k] * Wc[k * 10 + o];
  logits[t] = acc;
}

// Row softmax entropy + global max (entropy >= 0 -> uint bit ordering is valid)
__global__ void entropy_kernel(const float* __restrict__ logits,
                               float* __restrict__ Hent,
                               unsigned* __restrict__ maxbits, int rows) {
  const int r = blockIdx.x * blockDim.x + threadIdx.x;
  if (r >= rows) return;
  float l[10], mx = -3.0e38f;
  #pragma unroll
  for (int o = 0; o < 10; ++o) { l[o] = logits[r * 10 + o]; mx = fmaxf(mx, l[o]); }
  float s = 0.0f;
  #pragma unroll
  for (int o = 0; o < 10; ++o) { l[o] = __expf(l[o] - mx); s += l[o]; }
  const float inv = 1.0f / s;
  float ent = 0.0f;
  #pragma unroll
  for (int o = 0; o < 10; ++o) {
    const float p = l[o] * inv;
    ent -= p * __logf(fmaxf(p, 1e-30f));
  }
  Hent[r] = ent;
  atomicMax(maxbits, __float_as_uint(ent));
}

// pooled_h[b,:] = sum_c (1 - Hent[b*C+c]/maxH) * hp[b,c,:]
__global__ void lam_pool(const float* __restrict__ hp,
                         const float* __restrict__ Hent,
                         const unsigned* __restrict__ maxbits,
                         float* __restrict__ ph) {
  const int b = blockIdx.x, f = threadIdx.x;   // block = 256
  const float mh = __uint_as_float(*maxbits);
  float acc = 0.0f;
  for (int c = 0; c < NC; ++c) {
    const float lam = 1.0f - Hent[b * NC + c] / mh;
    acc += lam * hp[((size_t)b * NC + c) * H2D + f];
  }
  ph[(size_t)b * H2D + f] = acc;
}

// out[b,o] = pooled_h[b,:] @ cls_W[:,o] + cls_b[o]
__global__ void final_out_kernel(const float* __restrict__ ph,
                                 const float* __restrict__ Wc,
                                 const float* __restrict__ bc,
                                 float* __restrict__ out) {
  const int t = blockIdx.x * blockDim.x + threadIdx.x;
  if (t >= NG * 10) return;
  const int b = t / 10, o = t % 10;
  float acc = bc[o];
  for (int k = 0; k < H2D; ++k)
    acc += ph[(size_t)b * H2D + k] * Wc[k * 10 + o];
  out[t] = acc;
}

// ---------------------------------------------------------------------------
extern "C" void kernel_launch(void* const* d_in, const int* in_sizes, int n_in,
                              void* d_out, int out_size, void* d_ws, size_t ws_size,
                              hipStream_t stream) {
  const float* feat    = (const float*)d_in[0];
  const float* w_g     = (const float*)d_in[1];
  const float* w_m     = (const float*)d_in[2];
  const int*   src_g   = (const int*)  d_in[3];
  const int*   src_m   = (const int*)  d_in[4];
  const int*   labels  = (const int*)  d_in[6];
  const float* gcn_W1  = (const float*)d_in[7];
  const float* gcn_b1  = (const float*)d_in[8];
  const float* gcn_W2  = (const float*)d_in[9];
  const float* gcn_b2  = (const float*)d_in[10];
  const float* gin_eps = (const float*)d_in[11];
  const float* gin0_W1 = (const float*)d_in[12];
  const float* gin0_b1 = (const float*)d_in[13];
  const float* gin0_W2 = (const float*)d_in[14];
  const float* gin0_b2 = (const float*)d_in[15];
  const float* gin1_W1 = (const float*)d_in[16];
  const float* gin1_b1 = (const float*)d_in[17];
  const float* gin1_W2 = (const float*)d_in[18];
  const float* gin1_b2 = (const float*)d_in[19];
  const float* fin_W1  = (const float*)d_in[20];
  const float* fin_b1  = (const float*)d_in[21];
  const float* fin_W2  = (const float*)d_in[22];
  const float* fin_b2  = (const float*)d_in[23];
  const float* cls_W   = (const float*)d_in[24];
  const float* cls_b   = (const float*)d_in[25];

  const size_t SZ = (size_t)NTOT * HD;     // 6.4M floats per node buffer
  float* P0 = (float*)d_ws;
  float* P1 = P0 + SZ;
  float* P2 = P1 + SZ;
  float* P3 = P2 + SZ;
  float* stats = P3 + SZ;                  // 512 floats: [256 sums | 256 sqs]
  float* sums = stats, *sqs = stats + 256;

  const int GEMM_ROWS_NODE = NTOT / 16;    // 3125
  const int AGG_BLOCKS     = NTOT / 8;     // 6250
  const dim3 B256(256);

  auto bn_node = [&](const float* X, float* Y) {   // BN+ReLU over [NTOT, 128]
    zero_f32<<<2, B256, 0, stream>>>(stats, 512);
    bn_stats_kernel<<<(NTOT + 255) / 256, 128, 0, stream>>>(X, sums, sqs, NTOT, HD, 256);
    bn_apply_relu<<<(int)(SZ / 256), B256, 0, stream>>>(X, Y, sums, sqs, NTOT, HD);
  };

  // ---------------- GCN branch ----------------
  gemm_bias_wmma<<<dim3(GEMM_ROWS_NODE, 1), B256, 0, stream>>>(feat, gcn_W1, gcn_b1, P0, NTOT, HD, HD, 0);
  agg_gcn_kernel<<<AGG_BLOCKS, B256, 0, stream>>>(P0, w_g, src_g, P1);
  bn_node(P1, P0);
  gemm_bias_wmma<<<dim3(GEMM_ROWS_NODE, 1), B256, 0, stream>>>(P0, gcn_W2, gcn_b2, P1, NTOT, HD, HD, 0);
  agg_gcn_kernel<<<AGG_BLOCKS, B256, 0, stream>>>(P1, w_g, src_g, P2);
  bn_node(P2, P2);                                  // h_gcn -> P2

  // ---------------- GIN branch ----------------
  agg_gin_kernel<<<AGG_BLOCKS, B256, 0, stream>>>(feat, w_m, src_m, gin_eps, 0, P0);
  gemm_bias_wmma<<<dim3(GEMM_ROWS_NODE, 1), B256, 0, stream>>>(P0, gin0_W1, gin0_b1, P1, NTOT, HD, HD, 0);
  bn_node(P1, P1);
  gemm_bias_wmma<<<dim3(GEMM_ROWS_NODE, 1), B256, 0, stream>>>(P1, gin0_W2, gin0_b2, P0, NTOT, HD, HD, 1);
  bn_node(P0, P0);
  agg_gin_kernel<<<AGG_BLOCKS, B256, 0, stream>>>(P0, w_m, src_m, gin_eps, 1, P1);
  gemm_bias_wmma<<<dim3(GEMM_ROWS_NODE, 1), B256, 0, stream>>>(P1, gin1_W1, gin1_b1, P3, NTOT, HD, HD, 0);
  bn_node(P3, P3);
  gemm_bias_wmma<<<dim3(GEMM_ROWS_NODE, 1), B256, 0, stream>>>(P3, gin1_W2, gin1_b2, P1, NTOT, HD, HD, 1);
  bn_node(P1, P1);                                  // h_gin -> P1

  // ---------------- cluster pooling (reuse P0 region) ----------------
  const int POOL_N = NG * NC * H2D;                 // 819200
  float*    pooled = P0;
  float*    hp2    = P0 + POOL_N;
  float*    Amat   = P0 + 2 * POOL_N;               // 50*64*64 = 204800
  float*    logits = Amat + NG * NC * NC;           // 32000
  float*    Hent   = logits + NG * NC * 10;         // 3200
  float*    ph     = Hent + NG * NC;                // 12800
  unsigned* maxb   = (unsigned*)(ph + NG * H2D);

  zero_f32<<<(POOL_N + 255) / 256, B256, 0, stream>>>(pooled, POOL_N);
  pool_cluster<<<NTOT, 128, 0, stream>>>(P2, P1, labels, pooled);
  zero_f32<<<(NG * NC * NC + 255) / 256, B256, 0, stream>>>(Amat, NG * NC * NC);
  build_A<<<NEDGE / 256, B256, 0, stream>>>(src_g, labels, Amat);
  set_diag<<<NG, NC, 0, stream>>>(Amat);

  // ---------------- finalGCN on pooled graphs ----------------
  const int PROWS = NG * NC;                        // 3200
  auto bn_pool = [&](float* X) {                    // BN+ReLU over [3200, 256], in place
    zero_f32<<<2, B256, 0, stream>>>(stats, 512);
    bn_stats_kernel<<<PROWS / 64, H2D, 0, stream>>>(X, sums, sqs, PROWS, H2D, 64);
    bn_apply_relu<<<POOL_N / 256, B256, 0, stream>>>(X, X, sums, sqs, PROWS, H2D);
  };
  gemm_bias_wmma<<<dim3(PROWS / 16, 2), B256, 0, stream>>>(pooled, fin_W1, fin_b1, hp2, PROWS, H2D, H2D, 0);
  adj_mm<<<dim3(NG, NC), B256, 0, stream>>>(Amat, hp2, pooled);
  bn_pool(pooled);
  gemm_bias_wmma<<<dim3(PROWS / 16, 2), B256, 0, stream>>>(pooled, fin_W2, fin_b2, hp2, PROWS, H2D, H2D, 0);
  adj_mm<<<dim3(NG, NC), B256, 0, stream>>>(Amat, hp2, pooled);
  bn_pool(pooled);                                  // hflat = pooled

  // ---------------- entropy-weighted pooling + classify ----------------
  cls_logits<<<(PROWS * 10 + 255) / 256, B256, 0, stream>>>(pooled, cls_W, cls_b, logits, PROWS);
  zero_f32<<<1, 32, 0, stream>>>((float*)maxb, 1);
  entropy_kernel<<<(PROWS + 255) / 256, B256, 0, stream>>>(logits, Hent, maxb, PROWS);
  lam_pool<<<NG, H2D, 0, stream>>>(pooled, Hent, maxb, ph);
  final_out_kernel<<<(NG * 10 + 255) / 256, B256, 0, stream>>>(ph, cls_W, cls_b, (float*)d_out);
}